// DiGCNIB_43611097924212
// MI455X (gfx1250) — compile-verified
//
#include <hip/hip_runtime.h>
#include <hip/hip_bf16.h>

// ---------------------------------------------------------------------------
// DiGCN inception blocks on MI455X (gfx1250, wave32).
//   per block: out = x@Wl + (bl+ba+bb); h=x@Wa; out[dst]+=ew*h[src];
//              h=x@Wb; out[dst2]+=ew2*h[src2]
// GEMM: V_WMMA_F32_16X16X4_F32 (exact fp32), W staged to LDS once per
// workgroup via the Tensor Data Mover (TENSOR_LOAD_TO_LDS + s_wait_tensorcnt).
// Edge scatter: L2-side global_atomic_add_f32 (h/out fit in the 192MB L2).
// ---------------------------------------------------------------------------

typedef float        v2f __attribute__((ext_vector_type(2)));
typedef float        v8f __attribute__((ext_vector_type(8)));
typedef unsigned int v4u __attribute__((ext_vector_type(4)));
typedef int          v4i __attribute__((ext_vector_type(4)));
typedef int          v8i __attribute__((ext_vector_type(8)));

#define DCOL 128  // all feature dims are 128

// C[M x 128] = A[M x 128] * W[128 x 128] (+ optional bias b0+b1+b2 per col)
// One wave -> 16 rows x all 128 cols (8 WMMA tiles). blockDim = 256 (8 waves).
__global__ __launch_bounds__(256)
void digcn_gemm_wmma_f32(const float* __restrict__ A,
                         const float* __restrict__ W,
                         const float* __restrict__ b0,
                         const float* __restrict__ b1,
                         const float* __restrict__ b2,
                         float* __restrict__ C, int M) {
  __shared__ float sW[DCOL * DCOL];            // 64 KB: whole weight matrix

  const int lane = threadIdx.x & 31;
  const int wave = threadIdx.x >> 5;

  // ---- stage W into LDS ----------------------------------------------------
#if defined(__AMDGCN__) && __has_builtin(__builtin_amdgcn_tensor_load_to_lds)
  if (wave == 0) {
    const unsigned long long ga = (unsigned long long)(uintptr_t)W;
    const unsigned int lds = (unsigned int)(uintptr_t)sW;  // low 32b = LDS byte offset
    // D# group0: count=1 | lds_addr | global_addr[31:0] | global_addr[56:32], type=2
    v4u g0 = { 1u, lds, (unsigned int)(ga & 0xffffffffu),
               (unsigned int)(((ga >> 32) & 0x01ffffffu) | 0x80000000u) };
    // D# group1: data_size=4B; tensor_dim0=128; tensor_dim1=128; tile_dim0=128;
    //            tile_dim1=128; tensor_dim0_stride=128 (f32 elements)
    v8i g1 = { (int)(2u << 16),      // [17:16] data_size = 2 (4 bytes)
               (int)(128u << 16),    // tensor_dim0[15:0] -> bits [79:48] lo
               (int)(128u << 16),    // tensor_dim1[15:0] -> bits [111:80] lo
               (int)(128u << 16),    // tile_dim0 -> bits [127:112]
               (int)128,             // tile_dim1 -> bits [143:128]
               (int)128,             // tensor_dim0_stride[31:0]
               0, 0 };
    v4i gz4 = { 0, 0, 0, 0 };        // groups 2/3 unused (2-D tensor)
    v8i gz8 = { 0, 0, 0, 0, 0, 0, 0, 0 };
    __builtin_amdgcn_tensor_load_to_lds(g0, g1, gz4, gz4, gz8, 0);
    __builtin_amdgcn_s_wait_tensorcnt(0);
  }
#else
  for (int i = threadIdx.x; i < DCOL * DCOL / 4; i += 256) {
    ((float4*)sW)[i] = ((const float4*)W)[i];
  }
#endif
  __syncthreads();                    // before tail-wave exit: all waves arrive

  const int row0 = (blockIdx.x * 8 + wave) * 16;
  if (row0 >= M) return;              // wave-uniform exit

  const int half = lane >> 4;         // 0: lanes 0-15, 1: lanes 16-31
  const int l15  = lane & 15;

  v8f acc[8];
#pragma unroll
  for (int t = 0; t < 8; ++t) acc[t] = (v8f){0.f,0.f,0.f,0.f,0.f,0.f,0.f,0.f};

  // A fragment: lane holds A[row0+l15][k + 2*half + {0,1}] -> contiguous pair
  const float* arow = A + (size_t)(row0 + l15) * DCOL;

  for (int kk = 0; kk < DCOL; kk += 4) {
    const float2 ap = *(const float2*)(arow + kk + 2 * half);
    v2f a; a.x = ap.x; a.y = ap.y;
    const float* wr0 = &sW[(kk + 2 * half) * DCOL];   // row k (in LDS)
#pragma unroll
    for (int t = 0; t < 8; ++t) {
      const int col = t * 16 + l15;
      v2f b; b.x = wr0[col]; b.y = wr0[DCOL + col];   // rows k, k+1
      acc[t] = __builtin_amdgcn_wmma_f32_16x16x4_f32(
          /*neg_a=*/false, a, /*neg_b=*/false, b,
          /*c_mod=*/(short)0, acc[t], /*reuse_a=*/false, /*reuse_b=*/false);
    }
  }

  // D layout: lane l15 holds N=col; VGPR r holds M = row0 + 8*half + r
  const int mbase = row0 + half * 8;
  const bool has_bias = (b0 != nullptr);
#pragma unroll
  for (int t = 0; t < 8; ++t) {
    const int col = t * 16 + l15;
    const float bias = has_bias ? (b0[col] + b1[col] + b2[col]) : 0.0f;
#pragma unroll
    for (int r = 0; r < 8; ++r) {
      C[(size_t)(mbase + r) * DCOL + col] = acc[t][r] + bias;
    }
  }
}

// out[dst[e]][:] += ew[e] * h[src[e]][:]   (one wave per edge, float4 per lane,
// accumulation via L2 atomics; edge_index is int64: row0=src, row1=dst)
__global__ __launch_bounds__(256)
void digcn_scatter_edges(const float* __restrict__ h,
                         const long long* __restrict__ ei,
                         const float* __restrict__ ew,
                         float* __restrict__ out, int nE) {
  const long long t = (long long)blockIdx.x * 256 + threadIdx.x;
  const long long e = t >> 5;
  if (e >= nE) return;
  const int lane = (int)(t & 31);

  const long long src = ei[e];
  const long long dst = ei[(long long)nE + e];
  const float w = ew[e];

  const float4 v = *(const float4*)(h + src * DCOL + lane * 4);
  float* o = out + dst * DCOL + lane * 4;
  atomicAdd(o + 0, v.x * w);
  atomicAdd(o + 1, v.y * w);
  atomicAdd(o + 2, v.z * w);
  atomicAdd(o + 3, v.w * w);
}

extern "C" void kernel_launch(void* const* d_in, const int* in_sizes, int n_in,
                              void* d_out, int out_size, void* d_ws, size_t ws_size,
                              hipStream_t stream) {
  const float*     x   = (const float*)d_in[0];
  const long long* ei  = (const long long*)d_in[1];
  const float*     ew  = (const float*)d_in[2];
  const long long* ei2 = (const long long*)d_in[3];
  const float*     ew2 = (const float*)d_in[4];

  // dict order: for i in 1..3, tags l,a,b: W then b
  const float* Wl[3] = {(const float*)d_in[5],  (const float*)d_in[11], (const float*)d_in[17]};
  const float* bl[3] = {(const float*)d_in[6],  (const float*)d_in[12], (const float*)d_in[18]};
  const float* Wa[3] = {(const float*)d_in[7],  (const float*)d_in[13], (const float*)d_in[19]};
  const float* ba[3] = {(const float*)d_in[8],  (const float*)d_in[14], (const float*)d_in[20]};
  const float* Wb[3] = {(const float*)d_in[9],  (const float*)d_in[15], (const float*)d_in[21]};
  const float* bb[3] = {(const float*)d_in[10], (const float*)d_in[16], (const float*)d_in[22]};

  const int M = in_sizes[0] / DCOL;   // 100000 (multiple of 16)
  const int E = in_sizes[2];          // 3200000

  const size_t nodeBytes = (size_t)M * DCOL * sizeof(float);
  float* bufh = (float*)d_ws;                               // h scratch
  float* buf1 = (float*)((char*)d_ws + nodeBytes);          // block-1 out
  float* buf2 = (float*)((char*)d_ws + 2 * nodeBytes);      // block-2 out

  const int nWaves = M / 16;                                // 6250
  const dim3 gGemm((nWaves + 7) / 8);
  const dim3 gScat((unsigned)(((long long)E * 32 + 255) / 256));

  const float* xin = x;
  float* outs[3] = {buf1, buf2, (float*)d_out};
  for (int b = 0; b < 3; ++b) {
    float* o = outs[b];
    // out = x @ Wl + (bl+ba+bb)  -- full overwrite (handles poisoned buffers)
    digcn_gemm_wmma_f32<<<gGemm, 256, 0, stream>>>(xin, Wl[b], bl[b], ba[b], bb[b], o, M);
    // h = x @ Wa ; scatter over (ei, ew)
    digcn_gemm_wmma_f32<<<gGemm, 256, 0, stream>>>(xin, Wa[b], nullptr, nullptr, nullptr, bufh, M);
    digcn_scatter_edges<<<gScat, 256, 0, stream>>>(bufh, ei, ew, o, E);
    // h = x @ Wb ; scatter over (ei2, ew2)
    digcn_gemm_wmma_f32<<<gGemm, 256, 0, stream>>>(xin, Wb[b], nullptr, nullptr, nullptr, bufh, M);
    digcn_scatter_edges<<<gScat, 256, 0, stream>>>(bufh, ei2, ew2, o, E);
    xin = o;
  }
}